// vFM_cluster_3478923510125
// MI455X (gfx1250) — compile-verified
//
#include <hip/hip_runtime.h>
#include <stdint.h>

#define P_DIM 256
#define C_DIM 1000
#define N_DIM 512
#define ORDER_K 127   // P/2 - 1

typedef __attribute__((ext_vector_type(2))) float v2f;
typedef __attribute__((ext_vector_type(8))) float v8f;

// ---------- log(i0e(x)) : Abramowitz & Stegun 9.8.1 / 9.8.2 ----------
__device__ __forceinline__ double dev_log_i0e(double x) {
  if (x <= 3.75) {
    double t = x / 3.75; t *= t;
    double p = 1.0 + t*(3.5156229 + t*(3.0899424 + t*(1.2067492 +
               t*(0.2659732 + t*(0.0360768 + t*0.0045813)))));
    return log(p) - x;
  } else {
    double t = 3.75 / x;
    double p = 0.39894228 + t*(0.01328592 + t*(0.00225319 + t*(-0.00157565 +
               t*(0.00916281 + t*(-0.02057706 + t*(0.02635537 +
               t*(-0.01647633 + t*0.00392377)))))));
    return log(p) - 0.5*log(x);
  }
}

// ---------- Miller backward recurrence: returns log I_ORDER(kappa) - kappa ----------
// Per-element rescaling is mathematically identical to the reference's global
// any-isinf rescale (scale bookkeeping cancels in e0).
__device__ double miller_e0(double kappa) {
  double inv = 1.0 / kappa;
  double In = 1.0, In1 = 0.0;
  double scale = 0.0, scale0 = 0.0, E0 = 0.0;
#pragma unroll 1
  for (int i = 2*ORDER_K; i >= 1; --i) {
    double coef = (2.0*(double)i) * inv;
    double It = coef*In + In1;
    double nIn1;
    if (!(It < 1.0e300)) {            // overflow (or inf/nan) -> rescale
      double logI = log(In);
      It = coef + In1/In;
      nIn1 = 1.0;
      scale += logI;
      if (i >= ORDER_K) scale0 += logI;
    } else {
      nIn1 = In;
    }
    if (i == ORDER_K) E0 = nIn1;      // captured I_ORDER (scaled repr)
    In = It; In1 = nIn1;
  }
  return dev_log_i0e(kappa) + log(E0) - log(In) + scale0 - scale;
}

// ---------- K0: per-row squared norms of features ----------
__global__ void k_rownorm2(const float* __restrict__ F, float* __restrict__ Fn2) {
  __shared__ float red[256];
  int n = blockIdx.x, p = threadIdx.x;
  float v = F[n*P_DIM + p];
  red[p] = v*v;
  __syncthreads();
  for (int st = 128; st > 0; st >>= 1) {
    if (p < st) red[p] += red[p+st];
    __syncthreads();
  }
  if (p == 0) Fn2[n] = red[0];
}

// ---------- K1: update_CV + update_kappa (one block per class) ----------
__global__ void k_update_cv(const float* __restrict__ F, const int* __restrict__ labels,
                            const unsigned char* __restrict__ mask,
                            const float* __restrict__ Ave, const float* __restrict__ Amount,
                            float* __restrict__ AveN, float* __restrict__ kappa) {
  __shared__ float red[256];
  int c = blockIdx.x, p = threadIdx.x;
  float s = 0.f, cnt = 0.f;
  for (int n = 0; n < N_DIM; ++n) {
    if (mask[n] && labels[n] == c) { s += F[n*P_DIM + p]; cnt += 1.f; }
  }
  float denom = (cnt == 0.f) ? 1.f : cnt;
  float ave = s / denom;
  float w = cnt / (cnt + Amount[c]);
  if (!(w == w)) w = 0.f;             // NaN -> 0
  float anew = Ave[c*P_DIM + p]*(1.f - w) + ave*w;
  red[p] = anew*anew;
  __syncthreads();
  for (int st = 128; st > 0; st >>= 1) {
    if (p < st) red[p] += red[p+st];
    __syncthreads();
  }
  float R = sqrtf(red[0]);
  float kap = (float)P_DIM * R / (1.f - R*R);
  if (kap > 1e5f || kap < 0.f) kap = 1e5f;
  AveN[c*P_DIM + p] = anew / fmaxf(R, 1e-12f);
  if (p == 0) kappa[c] = kap;
}

// ---------- K2: logc per class ----------
__global__ void k_logc(const float* __restrict__ kappa, double* __restrict__ logc) {
  int c = blockIdx.x*blockDim.x + threadIdx.x;
  if (c >= C_DIM) return;
  double kap = (double)kappa[c];
  double e0 = miller_e0(kap);
  logc[c] = e0 + kap - (double)ORDER_K * log(kap + 1e-20);
}

// ---------- K3: G = F * AveN^T via V_WMMA_F32_16X16X4_F32, fused into kappa_new ----------
// One wave per 16x16 output tile. 64 WMMA steps of K=4 over P=256.
__global__ void k_gemm_kappa(const float* __restrict__ F, const float* __restrict__ AveN,
                             const float* __restrict__ kappa, const float* __restrict__ Fn2,
                             float* __restrict__ Kmat) {
  const int TILES_X = (C_DIM + 15) / 16;   // 63
  int wave = blockIdx.x * (blockDim.x >> 5) + (threadIdx.x >> 5);
  int lane = threadIdx.x & 31;
  int tx = wave % TILES_X;
  int ty = wave / TILES_X;
  if (ty >= N_DIM / 16) return;            // wave-uniform exit (EXEC stays all-ones)
  int m0 = ty * 16, n0 = tx * 16;
  int lm = lane & 15;
  int ks = (lane < 16) ? 0 : 2;            // A: lanes 0-15 hold K=0,1; lanes 16-31 hold K=2,3
  int nc  = n0 + lm;
  int ncl = (nc < C_DIM) ? nc : (C_DIM - 1);   // clamp tail-column loads

  const float* arow = F    + (size_t)(m0 + lm) * P_DIM + ks;
  const float* brow = AveN + (size_t)ncl       * P_DIM + ks;

  v8f acc = 0.0f;
#pragma unroll 4
  for (int k0 = 0; k0 < P_DIM; k0 += 4) {
    v2f a = *(const v2f*)(arow + k0);       // 8B aligned (ks even)
    v2f b = *(const v2f*)(brow + k0);
    acc = __builtin_amdgcn_wmma_f32_16x16x4_f32(false, a, false, b, (short)0, acc,
                                                false, false);
  }

  if (nc >= C_DIM) return;                  // after all WMMAs: divergence is safe now
  double kap = (double)kappa[ncl];
  int mbase = m0 + ((lane < 16) ? 0 : 8);   // C/D layout: VGPR j -> M=j (+8 for hi lanes)
#pragma unroll
  for (int j = 0; j < 8; ++j) {
    int m = mbase + j;
    double g = (double)acc[j];
    double kn2 = kap*kap + 2.0*kap*g + (double)Fn2[m];
    Kmat[(size_t)m * C_DIM + nc] = (float)sqrt(kn2 > 0.0 ? kn2 : 0.0);
  }
}

// ---------- K4: per-(n,c) Miller recurrence + logits + row L2-normalize ----------
__global__ void k_logits(const float* __restrict__ Kmat, const double* __restrict__ logc,
                         float* __restrict__ out) {
  __shared__ float red[256];
  int n = blockIdx.x, tid = threadIdx.x;
  float Lv[4]; int cs[4]; int cnt = 0;
  float ss = 0.f;
  for (int c = tid; c < C_DIM; c += 256) {
    double kn = (double)Kmat[(size_t)n * C_DIM + c];
    double e0 = miller_e0(kn);
    double L  = e0 + kn - (double)ORDER_K * log(kn + 1e-20) - logc[c];
    float Lf = (float)L;
    Lv[cnt] = Lf; cs[cnt] = c; ++cnt;
    ss += Lf * Lf;
  }
  red[tid] = ss;
  __syncthreads();
  for (int st = 128; st > 0; st >>= 1) {
    if (tid < st) red[tid] += red[tid+st];
    __syncthreads();
  }
  float inv = 1.f / fmaxf(sqrtf(red[0]), 1e-12f);
  for (int j = 0; j < cnt; ++j)
    out[(size_t)n * C_DIM + cs[j]] = Lv[j] * inv;
}

extern "C" void kernel_launch(void* const* d_in, const int* in_sizes, int n_in,
                              void* d_out, int out_size, void* d_ws, size_t ws_size,
                              hipStream_t stream) {
  const float*         F      = (const float*)d_in[0];          // (512,256) f32
  const int*           labels = (const int*)d_in[1];            // (512,) i32
  const unsigned char* mask   = (const unsigned char*)d_in[2];  // (512,) bool
  const float*         Ave    = (const float*)d_in[3];          // (1000,256) f32
  const float*         Amount = (const float*)d_in[4];          // (1000,) f32
  float* out = (float*)d_out;                                   // (512,1000) f32

  // workspace layout (8B-aligned first)
  char* ws = (char*)d_ws;
  double* logc  = (double*)ws;                       // 1000 f64
  float*  AveN  = (float*)(ws + 8192);               // 1000*256 f32
  float*  kappa = AveN + (size_t)C_DIM * P_DIM;      // 1000 f32
  float*  Fn2   = kappa + C_DIM;                     // 512 f32
  float*  Kmat  = Fn2 + N_DIM;                       // 512*1000 f32

  k_rownorm2 <<<N_DIM, 256, 0, stream>>>(F, Fn2);
  k_update_cv<<<C_DIM, 256, 0, stream>>>(F, labels, mask, Ave, Amount, AveN, kappa);
  k_logc     <<<(C_DIM + 255)/256, 256, 0, stream>>>(kappa, logc);

  const int TILES = (N_DIM/16) * ((C_DIM + 15)/16);  // 32 * 63 = 2016 waves
  k_gemm_kappa<<<TILES/8, 256, 0, stream>>>(F, AveN, kappa, Fn2, Kmat);  // 8 waves/block

  k_logits   <<<N_DIM, 256, 0, stream>>>(Kmat, logc, out);
}